// CodebookQTIPhyb_61057255080570
// MI455X (gfx1250) — compile-verified
//
#include <hip/hip_runtime.h>
#include <stdint.h>

// CodebookQTIPhyb: codes -> hash -> 512-entry float2 LUT gather with sign flip.
// Memory-bound streaming kernel: 128 MB in, 256 MB out (~16.5 us floor @ 23.3 TB/s).
// LUT (4 KB) staged in LDS via gfx1250 async global->LDS copy; gathers via ds_load_b64.

#define LUT_ENTRIES 512   // 1 << LUT_BITS, each entry = 2 floats (8 B) -> 4096 B

typedef __attribute__((ext_vector_type(4))) int   int4v;
typedef __attribute__((ext_vector_type(4))) float float4v;

// Address-space-qualified 16-byte vector types for the async-to-LDS builtin.
typedef __attribute__((address_space(1))) int4v gvec_t;  // global
typedef __attribute__((address_space(3))) int4v svec_t;  // LDS

__global__ __launch_bounds__(256) void
CodebookQTIPhyb_61057255080570_kernel(const int* __restrict__ codes,
                                      const float* __restrict__ lut,
                                      float* __restrict__ out,
                                      int n_codes) {
  __shared__ float2 lut_s[LUT_ENTRIES];  // 4 KB

  // ---- Stage the LUT into LDS: 256 threads x 16 B = 4096 B ----
#if __has_builtin(__builtin_amdgcn_global_load_async_to_lds_b128)
  {
    gvec_t* gsrc = (gvec_t*)lut + threadIdx.x;
    svec_t* ldst = (svec_t*)lut_s + threadIdx.x;
    __builtin_amdgcn_global_load_async_to_lds_b128(gsrc, ldst,
                                                   /*imm offset=*/0, /*cpol=*/0);
  }
#if __has_builtin(__builtin_amdgcn_s_wait_asynccnt)
  __builtin_amdgcn_s_wait_asynccnt(0);
#else
  asm volatile("s_wait_asynccnt 0" ::: "memory");
#endif
#else
  // Fallback: plain vectorized load + LDS store.
  {
    const float4v* src = (const float4v*)lut;
    ((float4v*)lut_s)[threadIdx.x] = src[threadIdx.x];
  }
#endif
  __syncthreads();

  // ---- Main streaming loop: 4 codes (16 B in, 32 B out) per thread ----
  const int n4 = n_codes >> 2;
  const int i = blockIdx.x * blockDim.x + threadIdx.x;
  if (i < n4) {
    const int4v c4 = __builtin_nontemporal_load((const int4v*)codes + i);

    float4v o0, o1;
#pragma unroll
    for (int j = 0; j < 4; ++j) {
      const uint32_t c    = (uint32_t)c4[j];
      const uint32_t t    = c * c + c;                 // (c+1)*c, uint32 wraparound exact
      const uint32_t sbit = (t & 0x8000u) << 16;       // bit15 -> float sign bit
      const float2   v    = lut_s[(t >> 6) & (LUT_ENTRIES - 1)];  // ds_load_b64 gather
      const float    vx   = __uint_as_float(__float_as_uint(v.x) ^ sbit);
      if (j < 2) { o0[2 * j] = vx; o0[2 * j + 1] = v.y; }
      else       { o1[2 * (j - 2)] = vx; o1[2 * (j - 2) + 1] = v.y; }
    }
    __builtin_nontemporal_store(o0, (float4v*)out + 2 * i);
    __builtin_nontemporal_store(o1, (float4v*)out + 2 * i + 1);
  }

  // ---- Scalar tail for n_codes not divisible by 4 ----
  const int tail = n_codes & 3;
  if (blockIdx.x == 0 && (int)threadIdx.x < tail) {
    const int      k    = (n_codes & ~3) + (int)threadIdx.x;
    const uint32_t c    = (uint32_t)codes[k];
    const uint32_t t    = c * c + c;
    const uint32_t sbit = (t & 0x8000u) << 16;
    const float2   v    = lut_s[(t >> 6) & (LUT_ENTRIES - 1)];
    out[2 * k]     = __uint_as_float(__float_as_uint(v.x) ^ sbit);
    out[2 * k + 1] = v.y;
  }
}

extern "C" void kernel_launch(void* const* d_in, const int* in_sizes, int n_in,
                              void* d_out, int out_size, void* d_ws, size_t ws_size,
                              hipStream_t stream) {
  const int*   codes = (const int*)d_in[0];    // [8192*4096] int32
  const float* lut   = (const float*)d_in[1];  // [512*2] float32
  float*       outp  = (float*)d_out;          // [8192*4096*2] float32

  const int n_codes = in_sizes[0];
  const int n4      = n_codes >> 2;
  const int threads = 256;
  int blocks        = (n4 + threads - 1) / threads;
  if (blocks < 1) blocks = 1;

  CodebookQTIPhyb_61057255080570_kernel<<<blocks, threads, 0, stream>>>(
      codes, lut, outp, n_codes);
}